// GATConvHP_67534065762525
// MI455X (gfx1250) — compile-verified
//
#include <hip/hip_runtime.h>

#define N_NODES 100000
#define N_EDGES 1600000
#define IN_F    256
#define HEADS   4
#define OUT_F   32
#define HD      128   // HEADS*OUT_F

typedef __attribute__((ext_vector_type(2))) float v2f;
typedef __attribute__((ext_vector_type(8))) float v8f;

// ---------------------------------------------------------------------------
// K1: ft[N,128] = feat[N,256] @ fc_w[128,256]^T   via V_WMMA_F32_16X16X4_F32
// Block = 16 output rows; 8 waves/block, wave w owns 16x16 tile at cols w*16.
// A-frag (16x4 f32): lane L feeds row M=L&15, VGPR0/1 = K=(koff,koff+1),
//   koff = 2*(L>>4).  B-frag (4x16): lane L feeds col N=L&15, same K pattern.
// C/D (8 VGPR f32): VGPR i = row  base + i + 8*(L>>4), col = L&15.
// ---------------------------------------------------------------------------
__global__ __launch_bounds__(256) void gat_gemm_wmma(
    const float* __restrict__ feat, const float* __restrict__ fcw,
    float* __restrict__ ft)
{
    const int lane = threadIdx.x & 31;
    const int wave = threadIdx.x >> 5;          // 0..7 -> column tile
    const int l15  = lane & 15;
    const int koff = (lane >> 4) << 1;          // 0 or 2
    const int rowA = blockIdx.x * 16 + l15;     // M row this lane feeds
    const int colB = wave * 16 + l15;           // N col this lane feeds

    const float* aptr = feat + (size_t)rowA * IN_F + koff;
    const float* bptr = fcw  + (size_t)colB * IN_F + koff;

    v8f acc = {};
#pragma unroll 8
    for (int k = 0; k < IN_F; k += 4) {
        v2f a = *(const v2f*)(aptr + k);   // 8B-aligned: even float index
        v2f b = *(const v2f*)(bptr + k);
        acc = __builtin_amdgcn_wmma_f32_16x16x4_f32(
            /*neg_a=*/false, a, /*neg_b=*/false, b,
            /*c_mod=*/(short)0, acc, /*reuse_a=*/false, /*reuse_b=*/false);
    }

    const int mbase = blockIdx.x * 16 + ((lane >> 4) << 3);
#pragma unroll
    for (int i = 0; i < 8; ++i)
        ft[(size_t)(mbase + i) * HD + (wave * 16 + l15)] = acc[i];
}

// ---------------------------------------------------------------------------
// K2: per (node,head): el/er dot products; init emax=-inf, asum=0, out=0
// ---------------------------------------------------------------------------
__global__ void gat_node_prep(
    const float* __restrict__ ft,
    const float* __restrict__ attn_l, const float* __restrict__ attn_r,
    float* __restrict__ el, float* __restrict__ er,
    float* __restrict__ emax, float* __restrict__ asum,
    float* __restrict__ out)
{
    int idx = blockIdx.x * blockDim.x + threadIdx.x;   // n*HEADS + h
    if (idx >= N_NODES * HEADS) return;
    int n = idx >> 2, h = idx & 3;

    const float* f  = ft + (size_t)n * HD + h * OUT_F;
    const float* al = attn_l + h * OUT_F;
    const float* ar = attn_r + h * OUT_F;
    float sl = 0.f, sr = 0.f;
#pragma unroll
    for (int j = 0; j < OUT_F; ++j) {
        float v = f[j];
        sl += v * al[j];
        sr += v * ar[j];
    }
    el[idx] = sl;
    er[idx] = sr;
    emax[idx] = __int_as_float((int)0xFF800000u);  // -inf
    asum[idx] = 0.f;

    float* o = out + (size_t)n * HD + h * OUT_F;
#pragma unroll
    for (int j = 0; j < OUT_F; ++j) o[j] = 0.f;
}

// float atomic max via monotone int/uint punning (valid with -inf init)
__device__ __forceinline__ void atomicMaxF32(float* addr, float val) {
    if (val >= 0.f)
        atomicMax((int*)addr, __float_as_int(val));
    else
        atomicMin((unsigned int*)addr, (unsigned int)__float_as_int(val));
}

__device__ __forceinline__ float edge_score(
    const float* el, const float* er, int s, int d, int h)
{
    float x = el[s * HEADS + h] + er[d * HEADS + h];
    return x > 0.f ? x : 0.2f * x;   // leaky_relu(0.2)
}

// ---------------------------------------------------------------------------
// K3: per (edge,head): segment max of scores into emax[dst]
// ---------------------------------------------------------------------------
__global__ void gat_edge_max(
    const int* __restrict__ src, const int* __restrict__ dst,
    const float* __restrict__ el, const float* __restrict__ er,
    float* __restrict__ emax)
{
    int idx = blockIdx.x * blockDim.x + threadIdx.x;
    if (idx >= N_EDGES * HEADS) return;
    int e = idx >> 2, h = idx & 3;
    int s = src[e], d = dst[e];
    atomicMaxF32(&emax[d * HEADS + h], edge_score(el, er, s, d, h));
}

// ---------------------------------------------------------------------------
// K4: per (edge,head): asum[dst] += exp(score - emax[dst])
// ---------------------------------------------------------------------------
__global__ void gat_edge_sum(
    const int* __restrict__ src, const int* __restrict__ dst,
    const float* __restrict__ el, const float* __restrict__ er,
    const float* __restrict__ emax, float* __restrict__ asum)
{
    int idx = blockIdx.x * blockDim.x + threadIdx.x;
    if (idx >= N_EDGES * HEADS) return;
    int e = idx >> 2, h = idx & 3;
    int s = src[e], d = dst[e];
    float x = edge_score(el, er, s, d, h);
    atomicAdd(&asum[d * HEADS + h], expf(x - emax[d * HEADS + h]));
}

// ---------------------------------------------------------------------------
// K5: wave per edge: out[dst] += a_h * ft[src]  (128 floats, coalesced)
// ---------------------------------------------------------------------------
__global__ __launch_bounds__(256) void gat_aggregate(
    const int* __restrict__ src, const int* __restrict__ dst,
    const float* __restrict__ ft,
    const float* __restrict__ el, const float* __restrict__ er,
    const float* __restrict__ emax, const float* __restrict__ asum,
    float* __restrict__ out)
{
    int gid  = blockIdx.x * blockDim.x + threadIdx.x;
    int e    = gid >> 5;
    int lane = threadIdx.x & 31;
    if (e >= N_EDGES) return;

    int s = src[e], d = dst[e];

    // every lane computes weight for head (lane&3); lanes 0..3 are broadcast src
    int hh = lane & 3;
    float x = edge_score(el, er, s, d, hh);
    float a = expf(x - emax[d * HEADS + hh]) / asum[d * HEADS + hh];

    const float* fsrc = ft  + (size_t)s * HD;
    float*       odst = out + (size_t)d * HD;
#pragma unroll
    for (int h = 0; h < HEADS; ++h) {
        float ah = __shfl(a, h, 32);
        int c = h * OUT_F + lane;               // lane covers 0..31 feats of head h
        atomicAdd(&odst[c], fsrc[c] * ah);
    }
}

// ---------------------------------------------------------------------------
extern "C" void kernel_launch(void* const* d_in, const int* in_sizes, int n_in,
                              void* d_out, int out_size, void* d_ws, size_t ws_size,
                              hipStream_t stream)
{
    const float* feat   = (const float*)d_in[0];
    const float* fcw    = (const float*)d_in[1];
    const float* attn_l = (const float*)d_in[2];
    const float* attn_r = (const float*)d_in[3];
    const int*   src    = (const int*)d_in[4];
    const int*   dst    = (const int*)d_in[5];
    float*       out    = (float*)d_out;

    // workspace layout (~57.6 MB total)
    char* ws = (char*)d_ws;
    float* ft   = (float*)ws; ws += (size_t)N_NODES * HD * sizeof(float);
    float* el   = (float*)ws; ws += (size_t)N_NODES * HEADS * sizeof(float);
    float* er   = (float*)ws; ws += (size_t)N_NODES * HEADS * sizeof(float);
    float* emax = (float*)ws; ws += (size_t)N_NODES * HEADS * sizeof(float);
    float* asum = (float*)ws; ws += (size_t)N_NODES * HEADS * sizeof(float);

    // K1: projection GEMM (N = 6250*16 exactly, HD = 8*16 exactly)
    gat_gemm_wmma<<<N_NODES / 16, 256, 0, stream>>>(feat, fcw, ft);

    // K2: logits + init
    int nh = N_NODES * HEADS;
    gat_node_prep<<<(nh + 255) / 256, 256, 0, stream>>>(
        ft, attn_l, attn_r, el, er, emax, asum, out);

    // K3/K4: edge softmax statistics
    int eh = N_EDGES * HEADS;
    gat_edge_max<<<(eh + 255) / 256, 256, 0, stream>>>(src, dst, el, er, emax);
    gat_edge_sum<<<(eh + 255) / 256, 256, 0, stream>>>(src, dst, el, er, emax, asum);

    // K5: weighted scatter aggregation (wave per edge)
    gat_aggregate<<<(N_EDGES * 32) / 256, 256, 0, stream>>>(
        src, dst, ft, el, er, emax, asum, out);
}